// NGCF_6064493822517
// MI455X (gfx1250) — compile-verified
//
#include <hip/hip_runtime.h>
#include <hip/hip_bf16.h>

#define N_USER 100000
#define N_ITEM 200000
#define NTOT   300000
#define NNZ    3200000
#define EMB    24
#define DHID   64
#define BATCH  512
#define CONCAT 216   // 24 + 64*3

typedef __attribute__((ext_vector_type(2))) float v2f;
typedef __attribute__((ext_vector_type(8))) float v8f;

// D = A(16x4 f32) * B(4x16 f32) + C(16x16 f32)  -> v_wmma_f32_16x16x4_f32
__device__ __forceinline__ v8f wmma4(v2f a, v2f b, v8f c) {
  return __builtin_amdgcn_wmma_f32_16x16x4_f32(false, a, false, b, (short)0, c,
                                               false, false);
}

// ---------------------------------------------------------------- init concat
__global__ void __launch_bounds__(256)
k_init_all_e(const float* __restrict__ ue, const float* __restrict__ ie,
             float* __restrict__ all_e) {
  long long tid = (long long)blockIdx.x * blockDim.x + threadIdx.x;
  if (tid >= (long long)NTOT * EMB) return;
  int row = (int)(tid / EMB);
  int c   = (int)(tid % EMB);
  float v = (row < N_USER) ? ue[(long long)row * EMB + c]
                           : ie[(long long)(row - N_USER) * EMB + c];
  all_e[(long long)row * CONCAT + c] = v;
}

// ---------------------------------------------------------------------- zero
__global__ void __launch_bounds__(256)
k_zero4(float4* __restrict__ p, long long n4) {
  long long tid = (long long)blockIdx.x * blockDim.x + threadIdx.x;
  if (tid < n4) p[tid] = make_float4(0.f, 0.f, 0.f, 0.f);
}

// ------------------------------------------------------------ SpMM (scatter)
// side[row] += val * ego[col]; thread = (edge, 4-feature chunk)
__global__ void __launch_bounds__(256)
k_spmm(const int* __restrict__ er, const int* __restrict__ ec,
       const float* __restrict__ ev, const float* __restrict__ ego,
       int estride, float* __restrict__ side, int dchunks) {
  long long tid = (long long)blockIdx.x * blockDim.x + threadIdx.x;
  if (tid >= (long long)NNZ * dchunks) return;
  int e = (int)(tid / dchunks);
  int c = (int)(tid % dchunks) * 4;
  float v = ev[e];
  int   s = ec[e];
  int   t = er[e];
  const float4 x = *(const float4*)(ego + (long long)s * estride + c);
  float* dst = side + (long long)t * DHID + c;
  atomicAdd(dst + 0, v * x.x);
  atomicAdd(dst + 1, v * x.y);
  atomicAdd(dst + 2, v * x.z);
  atomicAdd(dst + 3, v * x.w);
}

// -------------------------------------------- fused dense layer (WMMA, f32)
// one wave per 16-row tile:
//   o = leaky( side@Wg + bg + (ego*side)@Wb + bb )
//   ego_out = o ; all_e[:, out_off:out_off+64] = o / max(||o||,1e-12)
__global__ void __launch_bounds__(32)
k_dense(const float* __restrict__ side, const float* __restrict__ ego_in,
        int in_stride, int d_in,
        const float* __restrict__ Wg, const float* __restrict__ bg,
        const float* __restrict__ Wb, const float* __restrict__ bb,
        float* __restrict__ ego_out, float* __restrict__ all_e, int out_off) {
  const int lane = threadIdx.x;
  const int m  = lane & 15;   // A row / B column handled by this lane
  const int kh = lane >> 4;   // K half select (K = 2*kh + j)
  const long long rbase = (long long)blockIdx.x * 16;

  v8f cg[4] = {v8f{}, v8f{}, v8f{}, v8f{}};
  v8f cb[4] = {v8f{}, v8f{}, v8f{}, v8f{}};

  const float* srow = side   + (rbase + m) * DHID;
  const float* erow = ego_in + (rbase + m) * (long long)in_stride;

  for (int kk = 0; kk < d_in; kk += 4) {
    const int k0 = kk + 2 * kh;
    v2f a = *(const v2f*)(srow + k0);        // side tile row
    v2f e = *(const v2f*)(erow + k0);        // ego  tile row
    v2f p = a * e;                           // elementwise ego*side
    const float* wg0 = Wg + (long long)k0 * DHID;
    const float* wb0 = Wb + (long long)k0 * DHID;
#pragma unroll
    for (int t = 0; t < 4; ++t) {
      const int col = t * 16 + m;
      v2f bgv; bgv[0] = wg0[col]; bgv[1] = wg0[DHID + col];
      v2f bbv; bbv[0] = wb0[col]; bbv[1] = wb0[DHID + col];
      cg[t] = wmma4(a, bgv, cg[t]);
      cb[t] = wmma4(p, bbv, cb[t]);
    }
  }

  float bsum[4];
#pragma unroll
  for (int t = 0; t < 4; ++t) bsum[t] = bg[t * 16 + m] + bb[t * 16 + m];

#pragma unroll
  for (int v = 0; v < 8; ++v) {
    float o[4], s = 0.f;
#pragma unroll
    for (int t = 0; t < 4; ++t) {
      float x = cg[t][v] + cb[t][v] + bsum[t];
      x = (x > 0.f) ? x : 0.2f * x;          // leaky_relu(0.2)
      o[t] = x;
      s += x * x;
    }
    // reduce squared sum across the 16 lanes of this half (row M = v + 8*kh)
    s += __shfl_xor(s, 1);
    s += __shfl_xor(s, 2);
    s += __shfl_xor(s, 4);
    s += __shfl_xor(s, 8);
    const float scale = 1.0f / fmaxf(sqrtf(s), 1e-12f);
    const long long row = rbase + v + 8 * kh;
    float* eo = ego_out + row * DHID;
    float* ae = all_e + row * CONCAT + out_off;
#pragma unroll
    for (int t = 0; t < 4; ++t) {
      eo[t * 16 + m] = o[t];                 // un-normalized -> next layer
      ae[t * 16 + m] = o[t] * scale;         // normalized    -> concat
    }
  }
}

// -------------------------------------------------- scores = U @ I^T (WMMA)
__global__ void __launch_bounds__(32)
k_score(const int* __restrict__ users, const int* __restrict__ items,
        const float* __restrict__ all_e, float* __restrict__ out) {
  const int lane = threadIdx.x;
  const int m  = lane & 15;
  const int kh = lane >> 4;
  const int mt = blockIdx.y, nt = blockIdx.x;
  const float* urow = all_e + (long long)users[mt * 16 + m] * CONCAT;
  const float* irow = all_e + (long long)(N_USER + items[nt * 16 + m]) * CONCAT;
  v8f c = v8f{};
  for (int kk = 0; kk < CONCAT; kk += 4) {   // 216 = 54 * 4
    const int k0 = kk + 2 * kh;
    v2f a = *(const v2f*)(urow + k0);
    v2f b = *(const v2f*)(irow + k0);
    c = wmma4(a, b, c);
  }
#pragma unroll
  for (int v = 0; v < 8; ++v) {
    const int row = mt * 16 + v + 8 * kh;
    out[(long long)row * BATCH + nt * 16 + m] = c[v];
  }
}

// ---------------------------------------------------------------------------
extern "C" void kernel_launch(void* const* d_in, const int* in_sizes, int n_in,
                              void* d_out, int out_size, void* d_ws,
                              size_t ws_size, hipStream_t stream) {
  const int*   users     = (const int*)d_in[0];
  const int*   pos_items = (const int*)d_in[1];
  const int*   adj_row   = (const int*)d_in[2];
  const int*   adj_col   = (const int*)d_in[3];
  const float* adj_val   = (const float*)d_in[4];
  const float* user_emb  = (const float*)d_in[5];
  const float* item_emb  = (const float*)d_in[6];
  const float* Wg[3] = {(const float*)d_in[7],  (const float*)d_in[11], (const float*)d_in[15]};
  const float* bg[3] = {(const float*)d_in[8],  (const float*)d_in[12], (const float*)d_in[16]};
  const float* Wb[3] = {(const float*)d_in[9],  (const float*)d_in[13], (const float*)d_in[17]};
  const float* bb[3] = {(const float*)d_in[10], (const float*)d_in[14], (const float*)d_in[18]};

  float* all_e = (float*)d_ws;                               // [N,216]
  float* side  = all_e + (long long)NTOT * CONCAT;           // [N,64]
  float* ego0  = side  + (long long)NTOT * DHID;             // [N,64]
  float* ego1  = ego0  + (long long)NTOT * DHID;             // [N,64]

  {
    long long n = (long long)NTOT * EMB;
    k_init_all_e<<<(unsigned)((n + 255) / 256), 256, 0, stream>>>(user_emb,
                                                                  item_emb, all_e);
  }

  const float* ego_in = all_e;   // layer-0 input lives in all_e[:, :24]
  int in_stride = CONCAT;
  int d_in_dim  = EMB;
  float* ego_bufs[3] = {ego0, ego1, ego0};
  int    out_offs[3] = {EMB, EMB + DHID, EMB + 2 * DHID};

  for (int k = 0; k < 3; ++k) {
    long long n4 = (long long)NTOT * DHID / 4;
    k_zero4<<<(unsigned)((n4 + 255) / 256), 256, 0, stream>>>((float4*)side, n4);

    int dch = d_in_dim / 4;
    long long nt = (long long)NNZ * dch;
    k_spmm<<<(unsigned)((nt + 255) / 256), 256, 0, stream>>>(
        adj_row, adj_col, adj_val, ego_in, in_stride, side, dch);

    k_dense<<<NTOT / 16, 32, 0, stream>>>(side, ego_in, in_stride, d_in_dim,
                                          Wg[k], bg[k], Wb[k], bb[k],
                                          ego_bufs[k], all_e, out_offs[k]);
    ego_in = ego_bufs[k];
    in_stride = DHID;
    d_in_dim  = DHID;
  }

  dim3 g(BATCH / 16, BATCH / 16);
  k_score<<<g, 32, 0, stream>>>(users, pos_items, all_e, (float*)d_out);
}